// FocalLoss_45878840656015
// MI455X (gfx1250) — compile-verified
//
#include <hip/hip_runtime.h>
#include <stdint.h>

// ---------------------------------------------------------------------------
// RetinaNet focal loss for MI455X (gfx1250, wave32).
//
// Core trick: squared anchor/annotation distances as a GEMM on the WMMA unit.
//   dxy2[m,n] = nrm_a[m] + nrm_k[n] - 2*(x_m*x_n + y_m*y_n)
//   A (16x4 f32) rows  = ( x_a, y_a, nrm_a, 1 )
//   B (4x16 f32) cols  = (-2x_k, -2y_k, 1, nrm_k )
//   D = A x B  (V_WMMA_F32_16X16X4_F32) -> 16x16 tile of squared distances.
// K=64 annotations -> 4 fixed B tiles per wave, reused for every anchor tile.
// Per-lane 4-way min reduces K 64->16 candidates; a 16-entry LDS scan per
// anchor finishes min+argmin (ties are vanishingly unlikely with random fp).
// All threshold tests use squared distance (monotone), so sqrt is avoided.
// ---------------------------------------------------------------------------

typedef __attribute__((ext_vector_type(2))) float v2f;
typedef __attribute__((ext_vector_type(8))) float v8f;

#define ALPHA_C     0.95f
#define ONEM_ALPHA  0.05f
#define DAMP        0.7f
#define EPS_C       1e-4f
#define KANN        64
#define NCLS        4

__global__ __launch_bounds__(256)
void focal_main(const float* __restrict__ cls,    // [B,N,4]
                const float* __restrict__ reg,    // [B,N,3]
                const float* __restrict__ anch,   // [B,N,3]  (reference uses image 0 only)
                const float* __restrict__ ann,    // [B,64,4] (x,y,alpha,class)
                float* __restrict__ partials,     // [B*4][numBlocks]
                int N, int numBlocks)
{
    const int b    = blockIdx.y;
    const int bx   = blockIdx.x;
    const int tid  = threadIdx.x;
    const int lane = tid & 31;
    const int wave = tid >> 5;
    const int hf   = lane >> 4;   // which half of the wave (K-split for A/B)
    const int l16  = lane & 15;   // M / N index within a 16-wide tile

    __shared__ float s_ann[KANN * 4];      // raw annotation rows (1 KB)
    __shared__ uint2 s_pairs[8][32][16];   // per wave: [anchorLocal][col] = (d2, k)
    __shared__ float s_red[8][4];

    // --- stage annotations into LDS with CDNA5 async-to-LDS DMA (waves 0,1) ---
    if (tid < KANN) {
        uint32_t ldsoff = (uint32_t)(uintptr_t)(&s_ann[0]) + (uint32_t)tid * 16u;
        const float* gsrc = ann + ((size_t)b * KANN + (size_t)tid) * 4;
        asm volatile("global_load_async_to_lds_b128 %0, %1, off"
                     :: "v"(ldsoff), "v"(gsrc)
                     : "memory");
        asm volatile("s_wait_asynccnt 0" ::: "memory");
    }
    __syncthreads();

    // --- build the 4 fixed B tiles (K = 64 annotations) in registers ---
    // B 4x16 f32 layout: N = lane%16; half0 holds K rows {0,1}, half1 {2,3}.
    v2f Bt[4];
    #pragma unroll
    for (int t = 0; t < 4; ++t) {
        int k = t * 16 + l16;
        float kx = s_ann[k * 4 + 0];
        float ky = s_ann[k * 4 + 1];
        if (hf == 0) { Bt[t].x = -2.0f * kx;        Bt[t].y = -2.0f * ky; }
        else         { Bt[t].x = 1.0f;              Bt[t].y = kx * kx + ky * ky; }
    }

    // --- two A tiles (32 anchors per wave), 8 WMMAs total ---
    // A 16x4 f32 layout: M = lane%16; half0 holds K cols {0,1}, half1 {2,3}.
    const int anchorBase = bx * 256 + wave * 32;
    v8f D[2][4];
    #pragma unroll
    for (int tA = 0; tA < 2; ++tA) {
        int m = anchorBase + tA * 16 + l16;
        if (m > N - 1) m = N - 1;              // keep EXEC all-ones for WMMA
        float ax = anch[(size_t)3 * m + 0];
        float ay = anch[(size_t)3 * m + 1];
        v2f A;
        if (hf == 0) { A.x = ax;                A.y = ay;  }
        else         { A.x = ax * ax + ay * ay; A.y = 1.0f; }
        v8f z = {};
        #pragma unroll
        for (int t = 0; t < 4; ++t) {
            D[tA][t] = __builtin_amdgcn_wmma_f32_16x16x4_f32(
                false, A, false, Bt[t], (short)0, z, false, false);
        }
    }

    // --- per-lane min over the 4 B tiles (K 64 -> 16 candidates per row) ---
    // D 16x16 layout: per lane, reg j holds row m = j + 8*hf, col n = lane%16.
    #pragma unroll
    for (int tA = 0; tA < 2; ++tA) {
        #pragma unroll
        for (int j = 0; j < 8; ++j) {
            float bv = D[tA][0][j]; int bt = 0;
            float v1 = D[tA][1][j]; if (v1 < bv) { bv = v1; bt = 1; }
            float v2 = D[tA][2][j]; if (v2 < bv) { bv = v2; bt = 2; }
            float v3 = D[tA][3][j]; if (v3 < bv) { bv = v3; bt = 3; }
            int al = tA * 16 + j + 8 * hf;     // anchor index within the wave's 32
            uint2 pr;
            pr.x = __float_as_uint(bv);
            pr.y = (unsigned)(bt * 16 + l16);  // global annotation index
            s_pairs[wave][al][l16] = pr;
        }
    }
    __syncthreads();

    // --- finish min+argmin: lane l scans the 16 candidates for its anchor ---
    float bestd2; int bestk;
    {
        uint2 p0 = s_pairs[wave][lane][0];
        bestd2 = __uint_as_float(p0.x); bestk = (int)p0.y;
        #pragma unroll
        for (int c = 1; c < 16; ++c) {
            uint2 p = s_pairs[wave][lane][c];
            float v = __uint_as_float(p.x);
            if (v < bestd2) { bestd2 = v; bestk = (int)p.y; }
        }
    }

    // --- scalar epilogue: one anchor per lane ---
    const int a = anchorBase + lane;
    float cls_acc = 0.f, xy_acc = 0.f, ang_acc = 0.f, np_acc = 0.f;
    if (a < N) {
        float annx = s_ann[bestk * 4 + 0];
        float anny = s_ann[bestk * 4 + 1];
        float anna = s_ann[bestk * 4 + 2];
        int   annc = (int)s_ann[bestk * 4 + 3];
        float ax     = anch[(size_t)3 * a + 0];
        float ay     = anch[(size_t)3 * a + 1];
        float aalpha = anch[(size_t)3 * a + 2];
        float adist  = fabsf(aalpha - anna);

        // dxy_min >= 45 <=> d2 >= 2025 ; dxy_min < 30 <=> d2 < 900
        bool negm = (bestd2 >= 2025.0f) || (adist >= 30.0f);
        bool posm = (bestd2 <   900.0f) && (adist <  20.0f);

        // focal-weighted BCE over 4 classes (target: pos->onehot, neg->0, else -1)
        const float4 p4 = ((const float4*)cls)[(size_t)b * N + a];
        float pc[4] = { p4.x, p4.y, p4.z, p4.w };
        #pragma unroll
        for (int c = 0; c < NCLS; ++c) {
            float p = fminf(fmaxf(pc[c], EPS_C), 1.0f - EPS_C);
            float t = posm ? ((c == annc) ? 1.0f : 0.0f) : (negm ? 0.0f : -1.0f);
            if (t >= 0.0f) {
                float l = (t == 1.0f)
                    ? ALPHA_C    * (1.0f - p) * (1.0f - p) * (-logf(p))
                    : ONEM_ALPHA * p * p                    * (-logf(1.0f - p));
                cls_acc += DAMP * l;
            }
        }

        // regression smooth-L1 (xy) + hinged angle, positives only
        if (posm) {
            size_t rb = ((size_t)b * N + a) * 3;
            float rx = reg[rb + 0], ry = reg[rb + 1], ra = reg[rb + 2];
            float tx = annx - ax, ty = anny - ay, ta = anna - aalpha;
            float dx = fabsf(tx - rx), dy = fabsf(ty - ry);
            const float inv9 = 1.0f / 9.0f;
            float lx = (dx <= inv9) ? 4.5f * dx * dx : (dx - 0.5f * inv9);
            float ly = (dy <= inv9) ? 4.5f * dy * dy : (dy - 0.5f * inv9);
            xy_acc  = DAMP * (lx + ly);
            ang_acc = DAMP * fmaxf((fabsf(ta - ra) - 10.0f) * 0.2f, 0.0f);
            np_acc  = 1.0f;
        }
    }

    // --- block reduction (wave shfl tree + LDS across 8 waves) ---
    #pragma unroll
    for (int off = 16; off >= 1; off >>= 1) {
        cls_acc += __shfl_down(cls_acc, off, 32);
        xy_acc  += __shfl_down(xy_acc,  off, 32);
        ang_acc += __shfl_down(ang_acc, off, 32);
        np_acc  += __shfl_down(np_acc,  off, 32);
    }
    if (lane == 0) {
        s_red[wave][0] = cls_acc; s_red[wave][1] = xy_acc;
        s_red[wave][2] = ang_acc; s_red[wave][3] = np_acc;
    }
    __syncthreads();
    if (tid < 4) {
        float s = 0.f;
        #pragma unroll
        for (int w = 0; w < 8; ++w) s += s_red[w][tid];
        partials[(size_t)(b * 4 + tid) * numBlocks + bx] = s;
    }
}

// Deterministic fixed-order sum of per-block partials + final normalization.
__global__ void focal_finalize(const float* __restrict__ partials,
                               float* __restrict__ out,
                               int numBlocks, int B)
{
    __shared__ float sums[16];
    int t = threadIdx.x;
    if (t < 4 * B) {
        float s = 0.f;
        const float* p = partials + (size_t)t * numBlocks;
        for (int i = 0; i < numBlocks; ++i) s += p[i];
        sums[t] = s;
    }
    __syncthreads();
    if (t == 0) {
        float cl = 0.f, xl = 0.f, al = 0.f;
        for (int b = 0; b < B; ++b) {
            float np    = sums[b * 4 + 3];
            float denom = fmaxf(np, 1.0f);
            cl += sums[b * 4 + 0] / denom;
            if (np > 0.f) {
                xl += sums[b * 4 + 1] / (2.0f * denom);
                al += sums[b * 4 + 2] / denom;
            }
        }
        float invB = 1.0f / (float)B;
        out[0] = cl * invB;
        out[1] = xl * invB;
        out[2] = al * invB;
    }
}

extern "C" void kernel_launch(void* const* d_in, const int* in_sizes, int n_in,
                              void* d_out, int out_size, void* d_ws, size_t ws_size,
                              hipStream_t stream)
{
    const float* cls  = (const float*)d_in[0];   // [B,N,4]
    const float* reg  = (const float*)d_in[1];   // [B,N,3]
    const float* anch = (const float*)d_in[2];   // [B,N,3]
    const float* ann  = (const float*)d_in[3];   // [B,64,4]

    const int B = in_sizes[3] / (KANN * 4);      // 4
    const int N = in_sizes[0] / (B * NCLS);      // 300000
    const int numBlocks = (N + 255) / 256;       // 1172

    float* partials = (float*)d_ws;              // B*4*numBlocks floats (~75 KB)

    dim3 grid(numBlocks, B);
    focal_main<<<grid, 256, 0, stream>>>(cls, reg, anch, ann, partials, N, numBlocks);
    focal_finalize<<<1, 64, 0, stream>>>(partials, (float*)d_out, numBlocks, B);
}